// EMDLoss_61186104099519
// MI455X (gfx1250) — compile-verified
//
#include <hip/hip_runtime.h>
#include <hip/hip_bf16.h>
#include <math.h>

// ---------------- constants ----------------
#define COS_EPS  1e-8f
#define LOG_EPS  1e-12f
#define SINK_EPS 0.5f
#define SINK_ITERS 5
#define NB 64
#define NC 64
#define NHW 64

typedef __attribute__((ext_vector_type(2))) float v2f;
typedef __attribute__((ext_vector_type(8))) float v8f;

// ---- TDM availability (this toolchain: 6-arg builtin, clang-23 style) ----
#if defined(__has_builtin)
#if __has_builtin(__builtin_amdgcn_tensor_load_to_lds) && \
    __has_builtin(__builtin_amdgcn_s_wait_tensorcnt)
#define HAS_TDM 1
#endif
#endif

#ifdef HAS_TDM
typedef unsigned int v4u __attribute__((ext_vector_type(4)));
typedef int          v8i __attribute__((ext_vector_type(8)));
typedef int          v4i __attribute__((ext_vector_type(4)));

// 2D f32 tile DMA: global (dim0 x dim1, row stride = stride0 elems) -> LDS,
// with optional LDS row padding (pad_amount_field: -1 = none, 0 = +1 DWORD
// per 64 DWORDs, 1 = +2 DWORDs per 64 DWORDs). D# layout per
// CDNA5 ISA.
__device__ __forceinline__ void tdm_load_2d_f32(
    unsigned int lds_off, const float* gsrc,
    unsigned int dim0, unsigned int dim1,
    unsigned int tile0, unsigned int tile1,
    unsigned int stride0, int pad_amount_field) {
    unsigned long long ga = (unsigned long long)(uintptr_t)gsrc;
    v4u g0;
    g0[0] = 1u;                                   // count=1 (valid user D#)
    g0[1] = lds_off;                              // lds_addr (bytes)
    g0[2] = (unsigned int)ga;                     // global_addr[31:0]
    g0[3] = (unsigned int)((ga >> 32) & 0x1FFFFFFu) | (2u << 30); // [56:32] | type=2
    v8i g1;
    unsigned int w0 = (2u << 16);                 // data_size = 4 bytes
    if (pad_amount_field >= 0)
        w0 |= (1u << 20)                          // pad_enable
            | (5u << 22)                          // pad_interval: 64 DWORDs
            | ((unsigned int)pad_amount_field << 25); // pad_amount code
    g1[0] = (int)w0;                              // mask=0 (not in cluster)
    g1[1] = (int)((dim0 & 0xFFFFu) << 16);        // tensor_dim0[15:0] @ b63:48
    g1[2] = (int)((dim0 >> 16) | ((dim1 & 0xFFFFu) << 16)); // dim0[31:16] | dim1[15:0]
    g1[3] = (int)((dim1 >> 16) | (tile0 << 16));  // dim1[31:16] | tile_dim0
    g1[4] = (int)(tile1 & 0xFFFFu);               // tile_dim1 | tile_dim2=0
    g1[5] = (int)stride0;                         // tensor_dim0_stride[31:0]
    g1[6] = 0;                                    // stride0[47:32] | stride1[15:0]
    g1[7] = 0;                                    // stride1[47:16]
    v4i z4 = {0, 0, 0, 0};                        // 2D: groups 2/3 unused
    v8i z8 = {0, 0, 0, 0, 0, 0, 0, 0};            // extra group (6-arg toolchain)
    __builtin_amdgcn_tensor_load_to_lds(g0, g1, z4, z4, z8, 0);
}
#endif

// ---------------- kernel 0: zero the scalar output ----------------
__global__ void zero_out_kernel(float* out) {
    if (threadIdx.x == 0) out[0] = 0.0f;
}

// ---------------- kernel 1: per-row stats (norm, max, logsumexp) --
// One wave32 per row of 64 floats: 2 elements per lane, butterfly reduce.
// stats layout: [tensor(4)][row(4096)][4] = {norm, rowmax, lse, pad}
__global__ __launch_bounds__(256) void stats_kernel(
    const float* __restrict__ t0, const float* __restrict__ t1,
    const float* __restrict__ t2, const float* __restrict__ t3,
    float* __restrict__ stats) {
    int wave = threadIdx.x >> 5;
    int lane = threadIdx.x & 31;
    int row  = blockIdx.x * 8 + wave;        // 0..16383
    int tensor = row >> 12;
    int r      = row & 4095;
    const float* src = (tensor == 0) ? t0 : (tensor == 1) ? t1
                     : (tensor == 2) ? t2 : t3;
    const float* x = src + (size_t)r * NHW;

    float a0 = x[2 * lane + 0];
    float a1 = x[2 * lane + 1];
    float ssq = a0 * a0 + a1 * a1;
    float mx  = fmaxf(a0, a1);
    #pragma unroll
    for (int m = 16; m >= 1; m >>= 1) {
        ssq += __shfl_xor(ssq, m, 32);
        mx   = fmaxf(mx, __shfl_xor(mx, m, 32));
    }
    float se = __expf(a0 - mx) + __expf(a1 - mx);
    #pragma unroll
    for (int m = 16; m >= 1; m >>= 1) se += __shfl_xor(se, m, 32);

    if (lane == 0) {
        float* s = stats + (size_t)row * 4;
        s[0] = sqrtf(ssq);      // row L2 norm
        s[1] = mx;              // row max
        s[2] = __logf(se);      // log sum exp(x - max)
    }
}

// ---------------- kernel 2: WMMA cosine-cost per batch ------------
// One 512-thread block (16 waves) per batch b. Operands staged into LDS by
// TDM DMA (padded stride 66 via D# pad fields) when available; each wave
// computes a 16x16 tile of G = X1 * X2^T via 16 chained
// v_wmma_f32_16x16x4_f32. Then C = 1 - G*rn1[i]*rn2[j], row min/max
// normalized, stored.
__global__ __launch_bounds__(512) void cost_kernel(
    const float* __restrict__ x1g, const float* __restrict__ x2g,
    const float* __restrict__ stats, int s1off, int s2off,
    float* __restrict__ Cout) {
    __shared__ float X1[64][66];   // stride 66: conflict-free WMMA operand reads
    __shared__ float X2[64][66];
    __shared__ float CT[64][65];   // stride 65: conflict-free row/col passes
    __shared__ float rn1[64], rn2[64], lo[64], sc[64];

    int b   = blockIdx.x;
    int tid = threadIdx.x;

#ifdef HAS_TDM
    if (tid < 32) {   // wave 0 issues both tile DMAs (TDM is wave-level)
        tdm_load_2d_f32((unsigned int)(uintptr_t)&X1[0][0], x1g + (size_t)b * 4096,
                        64, 64, 64, 64, 64, /*+2 DWORD pad / 64*/ 1);
        tdm_load_2d_f32((unsigned int)(uintptr_t)&X2[0][0], x2g + (size_t)b * 4096,
                        64, 64, 64, 64, 64, 1);
    }
#else
    {   // fallback: float4 staging through VGPRs
        const float4* p1 = (const float4*)(x1g + (size_t)b * 4096);
        const float4* p2 = (const float4*)(x2g + (size_t)b * 4096);
        for (int i = tid; i < 1024; i += 512) {
            int row = i >> 4, c4 = (i & 15) * 4;
            float4 v = p1[i];
            X1[row][c4+0] = v.x; X1[row][c4+1] = v.y;
            X1[row][c4+2] = v.z; X1[row][c4+3] = v.w;
            v = p2[i];
            X2[row][c4+0] = v.x; X2[row][c4+1] = v.y;
            X2[row][c4+2] = v.z; X2[row][c4+3] = v.w;
        }
    }
#endif
    // overlap with DMA: reciprocal clamped norms
    if (tid < 64) {
        rn1[tid] = 1.0f / fmaxf(stats[(size_t)(s1off + b * 64 + tid) * 4 + 0], COS_EPS);
        rn2[tid] = 1.0f / fmaxf(stats[(size_t)(s2off + b * 64 + tid) * 4 + 0], COS_EPS);
    }
#ifdef HAS_TDM
    if (tid < 32) __builtin_amdgcn_s_wait_tensorcnt(0);
#endif
    __syncthreads();

    // WMMA: wave w owns tile (ti, tj); A = X1 rows, B = X2^T (B loads mirror
    // A loads with N in place of M).
    int wid  = tid >> 5;
    int lane = tid & 31;
    int ti = (wid >> 2) * 16;
    int tj = (wid & 3) * 16;
    int m     = lane & 15;           // A: M index / B: N index
    int khalf = (lane >> 4) * 2;     // lane-half selects K pair
    v8f acc = {};
    #pragma unroll
    for (int kk = 0; kk < 16; ++kk) {
        int k = kk * 4 + khalf;
        v2f a, bb;
        a[0]  = X1[ti + m][k];  a[1]  = X1[ti + m][k + 1];
        bb[0] = X2[tj + m][k];  bb[1] = X2[tj + m][k + 1];
        acc = __builtin_amdgcn_wmma_f32_16x16x4_f32(
            /*neg_a=*/false, a, /*neg_b=*/false, bb,
            /*c_mod=*/(short)0, acc, /*reuse_a=*/false, /*reuse_b=*/false);
    }

    // D layout: VGPR r -> M = r + 8*(lane>>4); N = lane&15
    int hi = (lane >> 4) * 8;
    #pragma unroll
    for (int r = 0; r < 8; ++r) {
        int row = ti + r + hi;
        int col = tj + (lane & 15);
        CT[row][col] = 1.0f - acc[r] * rn1[row] * rn2[col];
    }
    __syncthreads();

    // Row min/max normalization (axis=-1, per (b,i) row)
    if (tid < 64) {
        float mn = CT[tid][0], mx = mn;
        #pragma unroll 8
        for (int j = 1; j < 64; ++j) {
            float v = CT[tid][j];
            mn = fminf(mn, v); mx = fmaxf(mx, v);
        }
        lo[tid] = mn;
        sc[tid] = 1.0f / (mx - mn);
    }
    __syncthreads();

    float* dst = Cout + (size_t)b * 4096;
    for (int i = tid; i < 4096; i += 512) {
        int row = i >> 6, col = i & 63;
        dst[i] = (CT[row][col] - lo[row]) * sc[row];
    }
}

// ---------------- kernel 3: Sinkhorn per (b,i) tile ---------------
// blockIdx = i*64 + b (i-major so the 64 blocks sharing C[i] are L2-adjacent).
// 64 threads per block; cost tile DMA'd into LDS by TDM (padded stride 65)
// while the block computes log-softmax rows; Gauss-Seidel u/v updates with
// max-subtracted logsumexp; final sum(exp(M)*C) reduced + atomicAdd.
__global__ __launch_bounds__(64) void sinkhorn_kernel(
    const float* __restrict__ x1g, const float* __restrict__ x2g,
    const float* __restrict__ Cg, const float* __restrict__ stats,
    int s1off, int s2off, float* __restrict__ out) {
    __shared__ float Ct[64][65];
    __shared__ float u[64], v[64], lp1[64], lp2[64], red[64];

    int bid = blockIdx.x;
    int i = bid >> 6;
    int b = bid & 63;
    int t = threadIdx.x;

#ifdef HAS_TDM
    if (t < 32)   // wave 0 kicks off the 16KB tile DMA (pad +1 DWORD / row)
        tdm_load_2d_f32((unsigned int)(uintptr_t)&Ct[0][0], Cg + (size_t)i * 4096,
                        64, 64, 64, 64, 64, /*+1 DWORD pad*/ 0);
#else
    {
        const float4* cp = (const float4*)(Cg + (size_t)i * 4096);
        __builtin_prefetch(cp, 0, 1);
        for (int q = t; q < 1024; q += 64) {
            float4 vv = cp[q];
            int row = q >> 4, c4 = (q & 15) * 4;
            Ct[row][c4+0] = vv.x; Ct[row][c4+1] = vv.y;
            Ct[row][c4+2] = vv.z; Ct[row][c4+3] = vv.w;
        }
    }
#endif
    // overlap with DMA: log(softmax + 1e-12) for row (b,i) of x1 and x2
    {
        const float* s1 = stats + (size_t)(s1off + b * 64 + i) * 4;
        float xv = x1g[(size_t)(b * 64 + i) * 64 + t];
        lp1[t] = __logf(__expf(xv - s1[1] - s1[2]) + LOG_EPS);
        const float* s2 = stats + (size_t)(s2off + b * 64 + i) * 4;
        float xw = x2g[(size_t)(b * 64 + i) * 64 + t];
        lp2[t] = __logf(__expf(xw - s2[1] - s2[2]) + LOG_EPS);
        u[t] = 0.0f; v[t] = 0.0f;
    }
#ifdef HAS_TDM
    if (t < 32) __builtin_amdgcn_s_wait_tensorcnt(0);
#endif
    __syncthreads();

    const float inv_eps = 1.0f / SINK_EPS;
    for (int it = 0; it < SINK_ITERS; ++it) {
        // u update: thread t handles row j=t; reads v[] (barrier-fresh)
        {
            float uj = u[t];
            float mmax = -3.0e38f;
            #pragma unroll 8
            for (int k = 0; k < 64; ++k)
                mmax = fmaxf(mmax, (-Ct[t][k] + uj + v[k]) * inv_eps);
            float s = 0.0f;
            #pragma unroll 8
            for (int k = 0; k < 64; ++k)
                s += __expf((-Ct[t][k] + uj + v[k]) * inv_eps - mmax);
            u[t] = SINK_EPS * (lp1[t] - (__logf(s) + mmax)) + uj;
        }
        __syncthreads();
        // v update: thread t handles col k=t; uses freshly-updated u[]
        {
            float vk = v[t];
            float mmax = -3.0e38f;
            #pragma unroll 8
            for (int j = 0; j < 64; ++j)
                mmax = fmaxf(mmax, (-Ct[j][t] + u[j] + vk) * inv_eps);
            float s = 0.0f;
            #pragma unroll 8
            for (int j = 0; j < 64; ++j)
                s += __expf((-Ct[j][t] + u[j] + vk) * inv_eps - mmax);
            v[t] = SINK_EPS * (lp2[t] - (__logf(s) + mmax)) + vk;
        }
        __syncthreads();
    }

    // loss tile = sum_{j,k} exp(M) * C
    float acc = 0.0f;
    float uj = u[t];
    #pragma unroll 8
    for (int k = 0; k < 64; ++k) {
        float c = Ct[t][k];
        acc += __expf((-c + uj + v[k]) * inv_eps) * c;
    }
    red[t] = acc;
    __syncthreads();
    for (int s = 32; s >= 1; s >>= 1) {
        if (t < s) red[t] += red[t + s];
        __syncthreads();
    }
    if (t == 0) atomicAdd(out, red[0] * (1.0f / 8192.0f)); // /4096 per pair, /2 pairs
}

// ---------------- host-side launch --------------------------------
extern "C" void kernel_launch(void* const* d_in, const int* in_sizes, int n_in,
                              void* d_out, int out_size, void* d_ws, size_t ws_size,
                              hipStream_t stream) {
    const float* f1_0 = (const float*)d_in[0];
    const float* f1_1 = (const float*)d_in[1];
    const float* f2_0 = (const float*)d_in[2];
    const float* f2_1 = (const float*)d_in[3];
    float* out = (float*)d_out;

    // workspace: stats (4*4096*4 floats = 256KB), then two 1MB cost buffers
    float* stats = (float*)d_ws;
    float* C0 = (float*)((char*)d_ws + (size_t)(1 << 18));
    float* C1 = (float*)((char*)d_ws + (size_t)(1 << 18) + (size_t)(1 << 20));

    zero_out_kernel<<<1, 1, 0, stream>>>(out);
    stats_kernel<<<2048, 256, 0, stream>>>(f1_0, f1_1, f2_0, f2_1, stats);
    // stats row offsets follow d_in order: f1_0->0, f1_1->4096, f2_0->8192, f2_1->12288
    cost_kernel<<<64, 512, 0, stream>>>(f1_0, f2_0, stats, 0,    8192,  C0);
    cost_kernel<<<64, 512, 0, stream>>>(f1_1, f2_1, stats, 4096, 12288, C1);
    sinkhorn_kernel<<<4096, 64, 0, stream>>>(f1_0, f2_0, C0, stats, 0,    8192,  out);
    sinkhorn_kernel<<<4096, 64, 0, stream>>>(f1_1, f2_1, C1, stats, 4096, 12288, out);
}